// SemanticDecoder_28604482192086
// MI455X (gfx1250) — compile-verified
//
#include <hip/hip_runtime.h>

#define BATCH   2048
#define HDIM    64
#define DDIM    300
#define TILES_M (BATCH / 16)            // 128
#define TILES_N ((DDIM + 15) / 16)      // 19

typedef float v2f __attribute__((ext_vector_type(2)));
typedef float v4f __attribute__((ext_vector_type(4)));
typedef float v8f __attribute__((ext_vector_type(8)));

// ---------------------------------------------------------------------------
// Kernel 1: stream 737 MB of zeros into final[B, D, D] with non-temporal
// 128-bit stores (region >> 192MB L2, so NT is the right temporal hint).
// D*D = 90000 floats per sample -> total 184,320,000 floats, /4 = 46,080,000
// float4 chunks, 16B-aligned. Grid-stride keeps stores fully coalesced.
// ---------------------------------------------------------------------------
__global__ void semdec_fill_zero_kernel(v4f* __restrict__ out, long n4) {
    long idx    = (long)blockIdx.x * blockDim.x + threadIdx.x;
    long stride = (long)gridDim.x * blockDim.x;
    v4f z = {0.0f, 0.0f, 0.0f, 0.0f};
    for (long i = idx; i < n4; i += stride) {
        __builtin_nontemporal_store(z, out + i);
    }
}

// ---------------------------------------------------------------------------
// Kernel 2: WMMA f32 GEMM (16x16 tile per wave, K=64 via 16x V_WMMA_F32_16X16X4_F32),
// computing mean and s simultaneously (shared A operand), then:
//   diff[row,col]                 = labels[row,col] - (mean + mu_b[col])
//   final[row, col, col]          = 1 / (s + sigma_b[col])^2   (diagonal scatter)
// Launched after the zero-fill on the same stream.
// ---------------------------------------------------------------------------
__global__ void semdec_gemm_diag_kernel(
    const float* __restrict__ bmat,     // [B, 64]
    const float* __restrict__ labels,   // [B, 300]
    const float* __restrict__ mu_w,     // [300, 64]
    const float* __restrict__ mu_b,     // [300]
    const float* __restrict__ sigma_w,  // [300, 64]
    const float* __restrict__ sigma_b,  // [300]
    float* __restrict__ final_out,      // [B, 300, 300] (pre-zeroed)
    float* __restrict__ diff_out)       // [B, 300]
{
    const int lane  = threadIdx.x & 31;
    const int wave  = threadIdx.x >> 5;
    const int tile  = blockIdx.x * (blockDim.x >> 5) + wave;
    if (tile >= TILES_M * TILES_N) return;

    const int tileM = tile / TILES_N;
    const int tileN = tile % TILES_N;

    const int rowBase = tileM * 16;
    const int colBase = tileN * 16;

    const int n16   = lane & 15;        // M for A, N for B/C/D
    const int khalf = (lane >> 4) * 2;  // K sub-offset: lanes 0-15 -> {0,1}, 16-31 -> {2,3}

    const int col  = colBase + n16;
    const int colc = (col < DDIM) ? col : (DDIM - 1);   // clamp for weight/bias loads

    const float* aRow = bmat    + (rowBase + n16) * HDIM + khalf;
    const float* wMu  = mu_w    + colc * HDIM + khalf;   // weight row == B-matrix column
    const float* wSg  = sigma_w + colc * HDIM + khalf;

    v8f cMu = {0.f, 0.f, 0.f, 0.f, 0.f, 0.f, 0.f, 0.f};
    v8f cSg = {0.f, 0.f, 0.f, 0.f, 0.f, 0.f, 0.f, 0.f};

#pragma unroll
    for (int k = 0; k < HDIM; k += 4) {
        v2f a  = *(const v2f*)(aRow + k);
        v2f bm = *(const v2f*)(wMu + k);
        v2f bs = *(const v2f*)(wSg + k);
        // (neg_a, A, neg_b, B, c_mod, C, reuse_a, reuse_b)
        cMu = __builtin_amdgcn_wmma_f32_16x16x4_f32(false, a, false, bm,
                                                    (short)0, cMu, false, false);
        cSg = __builtin_amdgcn_wmma_f32_16x16x4_f32(false, a, false, bs,
                                                    (short)0, cSg, false, false);
    }

    const float biasMu = mu_b[colc];
    const float biasSg = sigma_b[colc];
    const bool  colOK  = (col < DDIM);

    // C/D layout: VGPR g holds M = g + 8*(lane/16), N = lane%16
    const int mBase = rowBase + (lane >> 4) * 8;

#pragma unroll
    for (int g = 0; g < 8; ++g) {
        const int   row  = mBase + g;
        const float mean = cMu[g] + biasMu;
        const float s    = cSg[g] + biasSg;
        const float inv  = 1.0f / (s * s);
        if (colOK) {
            diff_out[row * DDIM + col] = labels[row * DDIM + col] - mean;
            final_out[(size_t)row * (size_t)(DDIM * DDIM)
                      + (size_t)col * DDIM + col] = inv;
        }
    }
}

extern "C" void kernel_launch(void* const* d_in, const int* in_sizes, int n_in,
                              void* d_out, int out_size, void* d_ws, size_t ws_size,
                              hipStream_t stream) {
    (void)in_sizes; (void)n_in; (void)out_size; (void)d_ws; (void)ws_size;

    const float* bmat    = (const float*)d_in[0];  // [2048, 64]
    const float* labels  = (const float*)d_in[1];  // [2048, 300]
    const float* mu_w    = (const float*)d_in[2];  // [300, 64]
    const float* mu_b    = (const float*)d_in[3];  // [300]
    const float* sigma_w = (const float*)d_in[4];  // [300, 64]
    const float* sigma_b = (const float*)d_in[5];  // [300]

    float* final_out = (float*)d_out;                               // [2048,300,300]
    float* diff_out  = (float*)d_out + (size_t)BATCH * DDIM * DDIM; // [2048,300]

    // Pass 1: non-temporal zero-fill of the 737 MB diagonal-matrix output.
    const long n4 = (long)BATCH * DDIM * DDIM / 4;   // 46,080,000 float4 chunks
    semdec_fill_zero_kernel<<<4096, 256, 0, stream>>>((v4f*)final_out, n4);

    // Pass 2: WMMA GEMM + bias + diff + 1/var diagonal scatter.
    // 128*19 = 2432 tiles, 8 waves (one tile each) per 256-thread block.
    const int tiles  = TILES_M * TILES_N;
    const int blocks = (tiles + 7) / 8;              // 304
    semdec_gemm_diag_kernel<<<blocks, 256, 0, stream>>>(
        bmat, labels, mu_w, mu_b, sigma_w, sigma_b, final_out, diff_out);
}